// OptimizePose_21930103013785
// MI455X (gfx1250) — compile-verified
//
#include <hip/hip_runtime.h>
#include <math.h>

typedef float v2f __attribute__((ext_vector_type(2)));
typedef float v8f __attribute__((ext_vector_type(8)));

// ---------------------------------------------------------------------------
// Kernel 1: build the 200 pose matrices M[p] = [exp(R[p]) | T[p]; 0 0 0 1]
// Rodrigues with the identity K^2 = w w^T - |w|^2 I:
//   Rot = (1 - b*theta2) * I + a*K + b * (w w^T)
// matching the reference's  theta = sqrt(theta2 + 1e-12), b denom clip(1e-12).
// ---------------------------------------------------------------------------
__global__ void pose_build_kernel(const float* __restrict__ R,
                                  const float* __restrict__ T,
                                  float* __restrict__ M, int P) {
    int p = blockIdx.x * blockDim.x + threadIdx.x;
    if (p >= P) return;
    float wx = R[p * 3 + 0], wy = R[p * 3 + 1], wz = R[p * 3 + 2];
    float t2 = wx * wx + wy * wy + wz * wz;
    float th = sqrtf(t2 + 1e-12f);
    float a  = __sinf(th) / th;
    float b  = (1.0f - __cosf(th)) / fmaxf(t2, 1e-12f);
    float c0 = 1.0f - b * t2;

    float* m = M + (p << 4);
    m[0]  = c0 + b * wx * wx;  m[1]  = -a * wz + b * wx * wy;  m[2]  =  a * wy + b * wx * wz;  m[3]  = T[p * 3 + 0];
    m[4]  =  a * wz + b * wy * wx;  m[5]  = c0 + b * wy * wy;  m[6]  = -a * wx + b * wy * wz;  m[7]  = T[p * 3 + 1];
    m[8]  = -a * wy + b * wz * wx;  m[9]  =  a * wx + b * wz * wy;  m[10] = c0 + b * wz * wz;  m[11] = T[p * 3 + 2];
    m[12] = 0.0f; m[13] = 0.0f; m[14] = 0.0f; m[15] = 1.0f;
}

// ---------------------------------------------------------------------------
// Kernel 2: out[i] = M[n[i]] @ E[i], 4 elements per wave via one
// V_WMMA_F32_16X16X4_F32 with block-diagonal packing.
//
// A (16x4): rows 4b..4b+3 = M[n[base+b]]   (ISA layout: lane = row,
//           VGPR0 = K0 (lanes 0-15) / K2 (lanes 16-31), VGPR1 = K1 / K3)
// B (4x16): cols 4b..4b+3 = E[base+b]      (lane = col, VGPR0 = row0/row2,
//           VGPR1 = row1/row3 by lane half)
// D (16x16): diagonal 4x4 blocks are the results.
//   VGPR v, lanes 0-15  -> D[v][lane]
//   VGPR v, lanes 16-31 -> D[v+8][lane-16]
// ---------------------------------------------------------------------------
__global__ __launch_bounds__(256) void pose_apply_wmma(
    const int* __restrict__ n, const float* __restrict__ E,
    const float* __restrict__ M, float* __restrict__ out, int N) {
    const int lane = threadIdx.x & 31;
    const long long group =
        (long long)blockIdx.x * (blockDim.x >> 5) + (threadIdx.x >> 5);
    const long long base = group << 2;            // 4 elements per wave
    const long long fullGroups = (long long)N >> 2;

    if (group < fullGroups) {                     // wave-uniform: EXEC all-1s
        const int l15 = lane & 15;
        const int b   = l15 >> 2;                 // element within group
        const int rc  = l15 & 3;                  // A: row of M_b; B: col of E_b
        const bool hi = lane >= 16;

        const long long eb = base + b;
        const int pose = n[eb];

        // A operand: pose matrix rows
        const float* Mp = M + ((long long)pose << 4) + (rc << 2);
        v2f Aop;
        Aop.x = Mp[hi ? 2 : 0];                   // K0 / K2
        Aop.y = Mp[hi ? 3 : 1];                   // K1 / K3

        // B operand: E columns
        const float* Ep = E + (eb << 4) + rc;
        v2f Bop;
        Bop.x = Ep[(hi ? 2 : 0) << 2];            // row0 / row2
        Bop.y = Ep[(hi ? 3 : 1) << 2];            // row1 / row3

        v8f C = {};
        v8f D = __builtin_amdgcn_wmma_f32_16x16x4_f32(
            /*neg_a=*/false, Aop, /*neg_b=*/false, Bop,
            /*c_mod=*/(short)0, C, /*reuse_a=*/false, /*reuse_b=*/false);

        // Diagonal-block extraction:
        //  elem0: VGPR0-3 @ lanes 0-3    elem1: VGPR4-7 @ lanes 4-7
        //  elem2: VGPR0-3 @ lanes 24-27  elem3: VGPR4-7 @ lanes 28-31
        const bool active = (lane < 8) || (lane >= 24);
        if (active) {
            const int eg  = (lane >> 2) & 3;      // 0,1 (low) / 2,3 (high)
            const int col = lane & 3;
            const bool sel = (eg & 1) != 0;       // odd elements use VGPR4-7
            float* op = out + ((base + eg) << 4) + col;
            op[0]  = sel ? D[4] : D[0];
            op[4]  = sel ? D[5] : D[1];
            op[8]  = sel ? D[6] : D[2];
            op[12] = sel ? D[7] : D[3];
        }
    } else if (group == fullGroups) {
        // Scalar tail for N % 4 leftovers (never taken for N = 2^20).
        const int tail = N & 3;
        if (lane < tail) {
            const long long e = base + lane;
            const int pose = n[e];
            const float* Mp = M + ((long long)pose << 4);
            const float* Ep = E + (e << 4);
            float* op = out + (e << 4);
#pragma unroll
            for (int r = 0; r < 4; ++r)
#pragma unroll
                for (int c = 0; c < 4; ++c) {
                    float acc = 0.0f;
#pragma unroll
                    for (int k = 0; k < 4; ++k)
                        acc += Mp[r * 4 + k] * Ep[k * 4 + c];
                    op[r * 4 + c] = acc;
                }
        }
    }
}

extern "C" void kernel_launch(void* const* d_in, const int* in_sizes, int n_in,
                              void* d_out, int out_size, void* d_ws,
                              size_t ws_size, hipStream_t stream) {
    const int*   n = (const int*)d_in[0];
    const float* E = (const float*)d_in[1];
    const float* R = (const float*)d_in[2];
    const float* T = (const float*)d_in[3];
    float*     out = (float*)d_out;

    const int N = in_sizes[0];
    const int P = in_sizes[2] / 3;
    float* Mws = (float*)d_ws;                    // P * 16 floats (12.8 KB)

    pose_build_kernel<<<(P + 255) / 256, 256, 0, stream>>>(R, T, Mws, P);

    const long long fullGroups  = (long long)N >> 2;
    const long long wavesNeeded = fullGroups + ((N & 3) ? 1 : 0);
    const long long threads     = wavesNeeded * 32;
    const int block = 256;
    const long long blocks = (threads + block - 1) / block;
    pose_apply_wmma<<<(int)blocks, block, 0, stream>>>(n, E, Mws, out, N);
}